// GNN_13657996001676
// MI455X (gfx1250) — compile-verified
//
#include <hip/hip_runtime.h>
#include <hip/hip_bf16.h>
#include <math.h>

#define N_NODES_C 50000
#define N_EDGES_C 800000
#define IN_DIM_C  256
#define H1_C      256
#define H2_C      128
#define OUT_DIM_C 64
#define NCODE_C   16
#define NRL_C     3   // residual VQ stages per layer

typedef float v2f __attribute__((ext_vector_type(2)));
typedef float v8f __attribute__((ext_vector_type(8)));

__device__ __forceinline__ float wave_sum(float v) {
#pragma unroll
  for (int off = 16; off > 0; off >>= 1) v += __shfl_xor(v, off, 32);
  return v;
}

__device__ __forceinline__ float selu_f(float x) {
  const float a = 1.6732632423543772f, s = 1.0507009873554805f;
  return x > 0.0f ? s * x : s * a * (expf(x) - 1.0f);
}

// ---------------------------------------------------------------------------
// Degree / inverse-sqrt-degree
// ---------------------------------------------------------------------------
__global__ void deg_kernel(const int* __restrict__ col, float* __restrict__ deg, int E) {
  int i = blockIdx.x * blockDim.x + threadIdx.x;
  if (i < E) atomicAdd(&deg[col[i]], 1.0f);
}

__global__ void dinv_kernel(float* __restrict__ deg, int N) {
  int i = blockIdx.x * blockDim.x + threadIdx.x;
  if (i < N) deg[i] = rsqrtf(deg[i] + 1.0f);  // +1 = self loop; deg>=1 guaranteed
}

// ---------------------------------------------------------------------------
// FP32 WMMA GEMM, 2x2 register-blocked: one wave computes a 32x32 output tile
// as four 16x16 V_WMMA_F32_16X16X4_F32 accumulators. Per K-step (K+=4):
// 4 fragment loads -> 4 WMMAs, halving A and B re-read traffic vs 1x1.
// Fragment layouts (ISA 7.12.2, f32):
//   A 16x4 : lane m=lane&15, ks=2*(lane>>4): v0=X[m][ks], v1=X[m][ks+1]
//   B 4x16 : lane n=lane&15:                 v0=W[ks][n], v1=W[ks+1][n]
//   C/D    : VGPR r -> M = r + 8*(lane>>4), N = lane&15
// M may have a 16-row tail tile: clamp second A row (VALU select, EXEC stays
// all-ones as WMMA requires) and skip the second store pair (wave-uniform).
// Nc must be a multiple of 32; K a multiple of 4.
// ---------------------------------------------------------------------------
__global__ __launch_bounds__(256)
void gemm_wmma_f32_2x2(const float* __restrict__ X, const float* __restrict__ W,
                       float* __restrict__ Y, int M, int K, int Nc) {
  const int lane   = threadIdx.x & 31;
  const int wave   = (blockIdx.x * blockDim.x + threadIdx.x) >> 5;
  const int tilesN = Nc >> 5;               // 32-wide N blocks
  const int tilesM = (M + 31) >> 5;         // 32-tall M blocks (tail allowed)
  if (wave >= tilesM * tilesN) return;      // wave-uniform
  const int tM = wave / tilesN;
  const int tN = wave - tM * tilesN;
  const int mr = lane & 15;
  const int kh = (lane >> 4) << 1;          // 0 or 2

  const int rowA0 = tM * 32 + mr;
  int       rowA1 = tM * 32 + 16 + mr;
  const bool hasM1 = (tM * 32 + 16) < M;    // wave-uniform
  rowA1 = min(rowA1, M - 1);                // clamp loads in tail tile

  const float* xp0 = X + (size_t)rowA0 * K + kh;
  const float* xp1 = X + (size_t)rowA1 * K + kh;
  const int n0 = tN * 32 + mr;
  const float* wp0 = W + (size_t)kh * Nc + n0;
  const float* wp1 = wp0 + 16;

  v8f c00 = {0.f,0.f,0.f,0.f,0.f,0.f,0.f,0.f};
  v8f c01 = c00, c10 = c00, c11 = c00;

  for (int k = 0; k < K; k += 4) {
    v2f a0, a1, b0, b1;
    a0.x = xp0[0];  a0.y = xp0[1];
    a1.x = xp1[0];  a1.y = xp1[1];
    b0.x = wp0[0];  b0.y = wp0[Nc];
    b1.x = wp1[0];  b1.y = wp1[Nc];
    c00 = __builtin_amdgcn_wmma_f32_16x16x4_f32(false, a0, false, b0, (short)0, c00, false, false);
    c01 = __builtin_amdgcn_wmma_f32_16x16x4_f32(false, a0, false, b1, (short)0, c01, false, false);
    c10 = __builtin_amdgcn_wmma_f32_16x16x4_f32(false, a1, false, b0, (short)0, c10, false, false);
    c11 = __builtin_amdgcn_wmma_f32_16x16x4_f32(false, a1, false, b1, (short)0, c11, false, false);
    xp0 += 4;
    xp1 += 4;
    wp0 += (size_t)4 * Nc;
    wp1 += (size_t)4 * Nc;
  }

  const int col0     = tN * 32 + mr;
  const int rowBase0 = tM * 32 + ((lane >> 4) << 3);
  float* y0 = Y + (size_t)rowBase0 * Nc + col0;
#pragma unroll
  for (int r = 0; r < 8; ++r) {
    y0[(size_t)r * Nc]      = c00[r];
    y0[(size_t)r * Nc + 16] = c01[r];
  }
  if (hasM1) {
    float* y1 = y0 + (size_t)16 * Nc;
#pragma unroll
    for (int r = 0; r < 8; ++r) {
      y1[(size_t)r * Nc]      = c10[r];
      y1[(size_t)r * Nc + 16] = c11[r];
    }
  }
}

// ---------------------------------------------------------------------------
// Edge aggregation: out[col[e]] += h[row[e]] * dinv[row[e]]*dinv[col[e]]
// One thread per (edge, 4-feature chunk); chunk-major within edge => coalesced.
// ---------------------------------------------------------------------------
__global__ __launch_bounds__(256)
void agg_edges(const float* __restrict__ h, float* __restrict__ out,
               const int* __restrict__ row, const int* __restrict__ col,
               const float* __restrict__ dinv, int E, int D) {
  const int chunks = D >> 2;
  int idx = blockIdx.x * blockDim.x + threadIdx.x;
  if (idx >= E * chunks) return;
  int e = idx / chunks;
  int c = idx - e * chunks;
  int r  = row[e];
  int cl = col[e];
  float nrm = dinv[r] * dinv[cl];
  const float4 v = *reinterpret_cast<const float4*>(h + (size_t)r * D + (size_t)c * 4);
  float* o = out + (size_t)cl * D + (size_t)c * 4;
  atomicAdd(o + 0, v.x * nrm);
  atomicAdd(o + 1, v.y * nrm);
  atomicAdd(o + 2, v.z * nrm);
  atomicAdd(o + 3, v.w * nrm);
}

// ---------------------------------------------------------------------------
// Post: agg += self-loop (lin*dinv^2) + bias, optional SELU. In-place on agg.
// ---------------------------------------------------------------------------
template <int ACT>
__global__ __launch_bounds__(256)
void post_kernel(float* __restrict__ agg, const float* __restrict__ lin,
                 const float* __restrict__ dinv, const float* __restrict__ bias,
                 int N, int D) {
  long long i = (long long)blockIdx.x * blockDim.x + threadIdx.x;
  if (i >= (long long)N * D) return;
  int node = (int)(i / D);
  int j    = (int)(i - (long long)node * D);
  float di = dinv[node];
  float v  = agg[i] + lin[i] * di * di + bias[j];
  if (ACT) v = selu_f(v);
  agg[i] = v;
}

// ---------------------------------------------------------------------------
// Residual VQ: grid-stride, one wave per node iteration. Codebook
// (3 stages x 16 codes x D) is L2-normalized once per block into LDS; the
// fixed small grid means the 48KB table is read ~1250x, not 6250x.
// argmax of cosine sim == argmax of dot(residual, cbn_k) since ||residual||
// is a positive per-node scalar. Loss: 0.25 * mean((q - residual)^2) summed.
// ---------------------------------------------------------------------------
template <int D>
__global__ __launch_bounds__(256)
void rvq_kernel(const float* __restrict__ h, const float* __restrict__ cb,
                float* __restrict__ lossAcc, float* __restrict__ ids,
                int N, int group) {
  __shared__ float cbn[NRL_C * NCODE_C * D];
  __shared__ float invn[NRL_C * NCODE_C];
  const int tid = threadIdx.x;
  for (int i = tid; i < NRL_C * NCODE_C * D; i += blockDim.x) cbn[i] = cb[i];
  __syncthreads();
  if (tid < NRL_C * NCODE_C) {
    float s = 0.0f;
    for (int j = 0; j < D; ++j) { float v = cbn[tid * D + j]; s += v * v; }
    invn[tid] = 1.0f / fmaxf(sqrtf(s), 1e-12f);
  }
  __syncthreads();
  for (int i = tid; i < NRL_C * NCODE_C * D; i += blockDim.x) cbn[i] *= invn[i / D];
  __syncthreads();

  const int lane   = tid & 31;
  const int wave0  = (blockIdx.x * blockDim.x + tid) >> 5;
  const int nWaves = (gridDim.x * blockDim.x) >> 5;
  constexpr int P = D / 32;
  const float lscale = 0.25f / ((float)N * (float)D);

  for (int node = wave0; node < N; node += nWaves) {
    float res[P];
#pragma unroll
    for (int p = 0; p < P; ++p) res[p] = h[(size_t)node * D + p * 32 + lane];

    for (int l = 0; l < NRL_C; ++l) {
      float best = -3.4e38f;
      int   bi   = 0;
      for (int k = 0; k < NCODE_C; ++k) {
        const float* cr = &cbn[(l * NCODE_C + k) * D];
        float part = 0.0f;
#pragma unroll
        for (int p = 0; p < P; ++p) part += res[p] * cr[p * 32 + lane];
        part = wave_sum(part);        // identical on all lanes -> bi wave-uniform
        if (part > best) { best = part; bi = k; }
      }
      const float* q = &cbn[(l * NCODE_C + bi) * D];
      float sq = 0.0f;
#pragma unroll
      for (int p = 0; p < P; ++p) {
        float qv = q[p * 32 + lane];
        float d  = qv - res[p];
        sq += d * d;
        res[p] -= qv;
      }
      sq = wave_sum(sq);
      if (lane == 0) {
        atomicAdd(lossAcc, sq * lscale);
        ids[(size_t)node * 9 + group * 3 + l] = (float)bi;
      }
    }
  }
}

// ---------------------------------------------------------------------------
// Global sum of h3 (for h / sum(h))
// ---------------------------------------------------------------------------
__global__ __launch_bounds__(256)
void sum_all_kernel(const float* __restrict__ x, float* __restrict__ acc, int n) {
  __shared__ float sdata[8];
  int i = blockIdx.x * blockDim.x + threadIdx.x;
  int stride = gridDim.x * blockDim.x;
  float s = 0.0f;
  for (; i < n; i += stride) s += x[i];
  s = wave_sum(s);
  int lane = threadIdx.x & 31, w = threadIdx.x >> 5;
  if (lane == 0) sdata[w] = s;
  __syncthreads();
  if (threadIdx.x == 0) {
    float t = 0.0f;
    for (int k = 0; k < 8; ++k) t += sdata[k];
    atomicAdd(acc, t);
  }
}

// ---------------------------------------------------------------------------
// Finalize: h /= S; h = tanh(h)^2; row L2-normalize. One wave per node (D=64).
// ---------------------------------------------------------------------------
__global__ __launch_bounds__(256)
void finalize_kernel(const float* __restrict__ h, const float* __restrict__ sumS,
                     float* __restrict__ out, int N) {
  const int lane = threadIdx.x & 31;
  const int node = (blockIdx.x * blockDim.x + threadIdx.x) >> 5;
  if (node >= N) return;
  float inv_s = 1.0f / (*sumS);
  float v0 = h[(size_t)node * 64 + lane] * inv_s;
  float v1 = h[(size_t)node * 64 + 32 + lane] * inv_s;
  float t0 = tanhf(v0); t0 *= t0;
  float t1 = tanhf(v1); t1 *= t1;
  float s = wave_sum(t0 * t0 + t1 * t1);
  float inv = 1.0f / fmaxf(sqrtf(s), 1e-12f);
  out[(size_t)node * 64 + lane]      = t0 * inv;
  out[(size_t)node * 64 + 32 + lane] = t1 * inv;
}

// ---------------------------------------------------------------------------
// Host launcher
// ---------------------------------------------------------------------------
extern "C" void kernel_launch(void* const* d_in, const int* in_sizes, int n_in,
                              void* d_out, int out_size, void* d_ws, size_t ws_size,
                              hipStream_t stream) {
  const float* x   = (const float*)d_in[0];
  const int*   ei  = (const int*)d_in[1];
  const float* W1  = (const float*)d_in[2];
  const float* b1  = (const float*)d_in[3];
  const float* W2  = (const float*)d_in[4];
  const float* b2  = (const float*)d_in[5];
  const float* W3  = (const float*)d_in[6];
  const float* b3  = (const float*)d_in[7];
  const float* cb1 = (const float*)d_in[8];
  const float* cb2 = (const float*)d_in[9];
  const float* cb3 = (const float*)d_in[10];

  const int*   rowp = ei;             // edge_index[0]
  const int*   colp = ei + N_EDGES_C; // edge_index[1]

  // workspace layout (floats)
  float* bufA = (float*)d_ws;                         // [50000, 256] lin output
  float* bufB = bufA + (size_t)N_NODES_C * H1_C;      // [50000, 256] agg / h
  float* dinv = bufB + (size_t)N_NODES_C * H1_C;      // [50000]
  float* scal = dinv + N_NODES_C;                     // [0]=sum(h3)

  // output layout (floats): h [50000*64] | loss [1] | ids [50000*9]
  float* hout  = (float*)d_out;
  float* lossp = hout + (size_t)N_NODES_C * OUT_DIM_C;
  float* idsp  = lossp + 1;

  const int TPB = 256;
  const int RVQ_BLOCKS = 1250;  // 10000 waves; ~5 nodes per wave
  auto blocks = [](long long n, int t) { return (int)((n + t - 1) / t); };
  auto gemm_blocks = [](int M, int Nc) {
    long long waves = (long long)((M + 31) / 32) * (Nc / 32);
    return (int)((waves * 32 + 255) / 256);
  };

  // ---- degree / dinv ----
  hipMemsetAsync(dinv, 0, N_NODES_C * sizeof(float), stream);
  hipMemsetAsync(scal, 0, sizeof(float), stream);
  hipMemsetAsync(lossp, 0, sizeof(float), stream);
  deg_kernel<<<blocks(N_EDGES_C, TPB), TPB, 0, stream>>>(colp, dinv, N_EDGES_C);
  dinv_kernel<<<blocks(N_NODES_C, TPB), TPB, 0, stream>>>(dinv, N_NODES_C);

  // ---- layer 1: [50000,256] @ [256,256] ----
  {
    gemm_wmma_f32_2x2<<<gemm_blocks(N_NODES_C, H1_C), TPB, 0, stream>>>(
        x, W1, bufA, N_NODES_C, IN_DIM_C, H1_C);
    hipMemsetAsync(bufB, 0, (size_t)N_NODES_C * H1_C * sizeof(float), stream);
    agg_edges<<<blocks((long long)N_EDGES_C * (H1_C / 4), TPB), TPB, 0, stream>>>(
        bufA, bufB, rowp, colp, dinv, N_EDGES_C, H1_C);
    post_kernel<1><<<blocks((long long)N_NODES_C * H1_C, TPB), TPB, 0, stream>>>(
        bufB, bufA, dinv, b1, N_NODES_C, H1_C);
    rvq_kernel<H1_C><<<RVQ_BLOCKS, TPB, 0, stream>>>(bufB, cb1, lossp, idsp,
                                                     N_NODES_C, 0);
  }

  // ---- layer 2: [50000,256] @ [256,128] ----
  {
    gemm_wmma_f32_2x2<<<gemm_blocks(N_NODES_C, H2_C), TPB, 0, stream>>>(
        bufB, W2, bufA, N_NODES_C, H1_C, H2_C);
    hipMemsetAsync(bufB, 0, (size_t)N_NODES_C * H2_C * sizeof(float), stream);
    agg_edges<<<blocks((long long)N_EDGES_C * (H2_C / 4), TPB), TPB, 0, stream>>>(
        bufA, bufB, rowp, colp, dinv, N_EDGES_C, H2_C);
    post_kernel<1><<<blocks((long long)N_NODES_C * H2_C, TPB), TPB, 0, stream>>>(
        bufB, bufA, dinv, b2, N_NODES_C, H2_C);
    rvq_kernel<H2_C><<<RVQ_BLOCKS, TPB, 0, stream>>>(bufB, cb2, lossp, idsp,
                                                     N_NODES_C, 1);
  }

  // ---- layer 3: [50000,128] @ [128,64], no SELU ----
  {
    gemm_wmma_f32_2x2<<<gemm_blocks(N_NODES_C, OUT_DIM_C), TPB, 0, stream>>>(
        bufB, W3, bufA, N_NODES_C, H2_C, OUT_DIM_C);
    hipMemsetAsync(bufB, 0, (size_t)N_NODES_C * OUT_DIM_C * sizeof(float), stream);
    agg_edges<<<blocks((long long)N_EDGES_C * (OUT_DIM_C / 4), TPB), TPB, 0, stream>>>(
        bufA, bufB, rowp, colp, dinv, N_EDGES_C, OUT_DIM_C);
    post_kernel<0><<<blocks((long long)N_NODES_C * OUT_DIM_C, TPB), TPB, 0, stream>>>(
        bufB, bufA, dinv, b3, N_NODES_C, OUT_DIM_C);
    rvq_kernel<OUT_DIM_C><<<RVQ_BLOCKS, TPB, 0, stream>>>(bufB, cb3, lossp, idsp,
                                                          N_NODES_C, 2);
  }

  // ---- finalize ----
  sum_all_kernel<<<1024, TPB, 0, stream>>>(bufB, scal, N_NODES_C * OUT_DIM_C);
  finalize_kernel<<<blocks((long long)N_NODES_C * 32, TPB), TPB, 0, stream>>>(
      bufB, scal, hout, N_NODES_C);
}